// AdditiveAttention2_41094247088138
// MI455X (gfx1250) — compile-verified
//
#include <hip/hip_runtime.h>
#include <hip/hip_bf16.h>
#include <math.h>

// ---------------------------------------------------------------------------
// AdditiveAttention fused for MI455X (gfx1250, wave32, WMMA).
//   B=8, Q=256, K=256, D=256, H=256, DV=512
// Never materializes features[B,Q,K,H] (would be ~1GB HBM traffic @23.3TB/s
// ~46us); fused traffic is ~12MB of inputs with k_proj/values L2-resident.
// GEMMs (projections + attn@V) use V_WMMA_F32_16X16X4_F32 (full fp32).
// Score loop (134M tanh+FMA, the real bottleneck) runs on the trans/VALU pipe:
// v_tanh_f32 if available, else v_exp_f32 + v_rcp_f32.
// ---------------------------------------------------------------------------

typedef __attribute__((ext_vector_type(2))) float v2f;
typedef __attribute__((ext_vector_type(8))) float v8f;

#define BATCH 8
#define QLEN  256
#define KLEN  256
#define DIN   256
#define HID_  256
#define DVAL  512

// ---- fast transcendental helpers -----------------------------------------
__device__ __forceinline__ float fast_exp2(float x) {
#if __has_builtin(__builtin_amdgcn_exp2f)
  return __builtin_amdgcn_exp2f(x);        // v_exp_f32
#else
  return exp2f(x);
#endif
}
__device__ __forceinline__ float fast_rcp(float x) {
#if __has_builtin(__builtin_amdgcn_rcpf)
  return __builtin_amdgcn_rcpf(x);         // v_rcp_f32
#else
  return 1.0f / x;
#endif
}
__device__ __forceinline__ float fast_exp(float x) {
  return fast_exp2(x * 1.4426950408889634f);
}
__device__ __forceinline__ float fast_tanh(float x) {
#if defined(__gfx1250__) && __has_builtin(__builtin_amdgcn_tanhf)
  return __builtin_amdgcn_tanhf(x);        // v_tanh_f32 (CDNA5 trans op)
#else
  // tanh(x) = 1 - 2/(exp2(2x*log2e)+1); saturates correctly at +/-inf.
  float t = fast_exp2(2.885390081777927f * x);
  return 1.0f - 2.0f * fast_rcp(t + 1.0f);
#endif
}

// ---- WMMA f32 16x16x4 wrapper ---------------------------------------------
// A: 16x4 f32, 2 VGPRs/lane. Lane L (m=L%16, hi=L>=16):
//   a.x = A[m][k + (hi?2:0)], a.y = A[m][k + (hi?3:1)]
// B: 4x16 f32, 2 VGPRs/lane (row striped across lanes):
//   b.x = B[k + (hi?2:0)][n], b.y = B[k + (hi?3:1)][n]   (n = L%16)
// C/D: 8 VGPRs; VGPR r -> row (hi?8+r:r), col n.
__device__ __forceinline__ v8f wmma_f32_16x16x4(v2f a, v2f b, v8f c) {
#if __has_builtin(__builtin_amdgcn_wmma_f32_16x16x4_f32)
  return __builtin_amdgcn_wmma_f32_16x16x4_f32(
      /*neg_a=*/false, a, /*neg_b=*/false, b,
      /*c_mod=*/(short)0, c, /*reuse_a=*/false, /*reuse_b=*/false);
#else
  // Compile-safety fallback only; not expected to be taken on gfx1250.
  v8f d = c;
  d[0] += a.x * b.x;
  return d;
#endif
}

// Coalesced stage of a 16x256 f32 tile (row stride `srcStride`) into padded
// LDS [16][257]. 256 threads, global_load_b128 (16B/lane), scalar LDS writes
// (row stride 257 is odd => conflict-free reads, but not 16B-aligned).
template <typename LDS>
__device__ __forceinline__ void stage_tile_16x256(LDS& lds, const float* __restrict__ src,
                                                  size_t srcStride, int tid) {
  for (int i = tid; i < 16 * 64; i += 256) {      // 1024 float4, 4 sweeps
    const int r = i >> 6, c4 = (i & 63) << 2;
    const float4 v = *reinterpret_cast<const float4*>(&src[(size_t)r * srcStride + c4]);
    lds[r][c4 + 0] = v.x;
    lds[r][c4 + 1] = v.y;
    lds[r][c4 + 2] = v.z;
    lds[r][c4 + 3] = v.w;
  }
}

// ---------------------------------------------------------------------------
// Kernel 1: k_proj[b*K + k][h] = keys[b,k,:] @ W_k[:,h]
// 2048 rows total; one 16-row tile per block; 8 waves x 2 col-tiles (16 cols).
// ---------------------------------------------------------------------------
__global__ void __launch_bounds__(256)
kproj_kernel(const float* __restrict__ keys, const float* __restrict__ Wk,
             float* __restrict__ kproj) {
  __shared__ float x_lds[16][257];   // padded: bank-conflict-free A-frag reads
  const int tid  = threadIdx.x;
  const int row0 = blockIdx.x * 16;

  stage_tile_16x256(x_lds, &keys[(size_t)row0 * DIN], DIN, tid);
  __syncthreads();

  const int wave = tid >> 5;
  const int lane = tid & 31;
  const int m    = lane & 15;
  const int hi   = lane >> 4;

  for (int t = 0; t < 2; ++t) {
    const int c0 = (wave * 2 + t) * 16;
    v8f acc = {};
    for (int k = 0; k < DIN; k += 4) {
      v2f a, b;
      a.x = x_lds[m][k + (hi ? 2 : 0)];
      a.y = x_lds[m][k + (hi ? 3 : 1)];
      b.x = Wk[(size_t)(k + (hi ? 2 : 0)) * HID_ + c0 + m];
      b.y = Wk[(size_t)(k + (hi ? 3 : 1)) * HID_ + c0 + m];
      acc = wmma_f32_16x16x4(a, b, acc);
    }
    for (int r = 0; r < 8; ++r)
      kproj[(size_t)(row0 + r + (hi ? 8 : 0)) * HID_ + c0 + m] = acc[r];
  }
}

// ---------------------------------------------------------------------------
// Kernel 2: per (b, 16-query tile):
//   P0: stage query tile -> k_lds (coalesced; k_lds dead until P2)
//   P1: q_lds = queries_tile @ W_q            (WMMA, straight into LDS)
//   P2: s[q][k] = sum_h wv[h]*tanh(q+k)       (VALU; wv via SMEM scalar loads)
//   P3: softmax rows                          (v_exp_f32)
//   P4: out_tile = attn @ values              (WMMA)
// ---------------------------------------------------------------------------
__global__ void __launch_bounds__(256)
attn_kernel(const float* __restrict__ queries, const float* __restrict__ values,
            const float* __restrict__ Wq, const float* __restrict__ wv,
            const float* __restrict__ kproj, float* __restrict__ out) {
  __shared__ float q_lds[16][257];
  __shared__ float k_lds[16][257];
  __shared__ float s_lds[16][257];
  __shared__ float red[32];          // [0..15]=row max, [16..31]=1/sum

  const int tid = threadIdx.x;
  const int b   = blockIdx.x >> 4;   // 16 q-tiles per batch
  const int q0  = (blockIdx.x & 15) * 16;

  const float* Qb  = queries + (size_t)b * QLEN * DIN;
  const float* Vb  = values  + (size_t)b * KLEN * DVAL;
  const float* KPb = kproj   + (size_t)b * KLEN * HID_;

  const int wave = tid >> 5;
  const int lane = tid & 31;
  const int m    = lane & 15;
  const int hi   = lane >> 4;

  // ---- Phase 0: stage query tile into k_lds (coalesced b128 loads) ----
  stage_tile_16x256(k_lds, &Qb[(size_t)q0 * DIN], DIN, tid);
  __syncthreads();

  // ---- Phase 1: q projection into LDS (2 col-tiles per wave) ----
  for (int t = 0; t < 2; ++t) {
    const int c0 = (wave * 2 + t) * 16;
    v8f acc = {};
    for (int k = 0; k < DIN; k += 4) {
      v2f a, bb;
      a.x  = k_lds[m][k + (hi ? 2 : 0)];
      a.y  = k_lds[m][k + (hi ? 3 : 1)];
      bb.x = Wq[(size_t)(k + (hi ? 2 : 0)) * HID_ + c0 + m];
      bb.y = Wq[(size_t)(k + (hi ? 3 : 1)) * HID_ + c0 + m];
      acc = wmma_f32_16x16x4(a, bb, acc);
    }
    for (int r = 0; r < 8; ++r)
      q_lds[r + (hi ? 8 : 0)][c0 + m] = acc[r];
  }

  // ---- Phase 2: scores. thread -> (q = tid/16, kc = tid%16); 16-k chunks ----
  // Per h: wv[h] is wave-uniform -> s_load (SMEM, no LDS traffic);
  // q_lds[qi][h] is a 2-address broadcast; k_lds[kc][h] is conflict-free.
  const int qi = tid >> 4;
  const int kc = tid & 15;
  for (int chunk = 0; chunk < 16; ++chunk) {
    __syncthreads();                 // protect q_lds/k_lds readers
    stage_tile_16x256(k_lds, &KPb[(size_t)(chunk * 16) * HID_], HID_, tid);
    if (chunk < 15)                  // global_prefetch_b8 of next chunk (16KB)
      __builtin_prefetch(&KPb[(size_t)(chunk + 1) * 16 * HID_ + tid * 16], 0, 3);
    __syncthreads();

    float acc = 0.f;
    for (int h = 0; h < HID_; ++h)
      acc = fmaf(wv[h], fast_tanh(q_lds[qi][h] + k_lds[kc][h]), acc);
    s_lds[qi][chunk * 16 + kc] = acc;
  }
  __syncthreads();

  // ---- Phase 3: softmax over K per row ----
  if (tid < 16) {
    float mx = -INFINITY;
    for (int k = 0; k < KLEN; ++k) mx = fmaxf(mx, s_lds[tid][k]);
    float sum = 0.f;
    for (int k = 0; k < KLEN; ++k) sum += fast_exp(s_lds[tid][k] - mx);
    red[tid]      = mx;
    red[16 + tid] = 1.0f / sum;
  }
  __syncthreads();
  {
    const float mx = red[qi], inv = red[16 + qi];
    for (int j = 0; j < 16; ++j) {
      const int k = kc + 16 * j;
      s_lds[qi][k] = fast_exp(s_lds[qi][k] - mx) * inv;
    }
  }
  __syncthreads();

  // ---- Phase 4: out[16,512] = attn[16,256] @ V[256,512]; 4 tiles per wave ----
  float* Ob = out + ((size_t)b * QLEN + q0) * DVAL;
  for (int t = 0; t < 4; ++t) {
    const int n0 = wave * 64 + t * 16;
    v8f acc = {};
    for (int k = 0; k < KLEN; k += 4) {
      v2f a, bb;
      a.x  = s_lds[m][k + (hi ? 2 : 0)];
      a.y  = s_lds[m][k + (hi ? 3 : 1)];
      bb.x = Vb[(size_t)(k + (hi ? 2 : 0)) * DVAL + n0 + m];
      bb.y = Vb[(size_t)(k + (hi ? 3 : 1)) * DVAL + n0 + m];
      acc = wmma_f32_16x16x4(a, bb, acc);
    }
    for (int r = 0; r < 8; ++r)
      Ob[(size_t)(r + (hi ? 8 : 0)) * DVAL + n0 + m] = acc[r];
  }
}

// ---------------------------------------------------------------------------
extern "C" void kernel_launch(void* const* d_in, const int* in_sizes, int n_in,
                              void* d_out, int out_size, void* d_ws, size_t ws_size,
                              hipStream_t stream) {
  const float* queries = (const float*)d_in[0];  // [8,256,256]
  const float* keys    = (const float*)d_in[1];  // [8,256,256]
  const float* values  = (const float*)d_in[2];  // [8,256,512]
  const float* Wq      = (const float*)d_in[3];  // [256,256] (in,out)
  const float* Wk      = (const float*)d_in[4];  // [256,256]
  const float* wvv     = (const float*)d_in[5];  // [256]
  float* out   = (float*)d_out;                  // [8,256,512]
  float* kproj = (float*)d_ws;                   // 2048*256 f32 = 2MB scratch

  // k_proj for all (b,k) rows: 2048/16 = 128 tile-blocks.
  kproj_kernel<<<128, 256, 0, stream>>>(keys, Wk, kproj);
  // fused q-proj + scores + softmax + attn@V: one block per (b, q-tile).
  attn_kernel<<<BATCH * (QLEN / 16), 256, 0, stream>>>(queries, values, Wq, wvv,
                                                       kproj, out);
}